// StandardAngular_38165079392278
// MI455X (gfx1250) — compile-verified
//
#include <hip/hip_runtime.h>

typedef float v4f __attribute__((ext_vector_type(4)));
typedef int   v4i __attribute__((ext_vector_type(4)));

#define LDS_ROW 36   // 32 data floats + 4 pad; 144B rows keep b128 LDS writes 16B-aligned
                     // and bank-conflict-free (each of 64 banks hit exactly twice/instr)

#if __has_builtin(__builtin_amdgcn_global_store_async_from_lds_b128)
#define USE_ASYNC_STORE 1
#else
#define USE_ASYNC_STORE 0
#endif

// Compute the 32 outputs for pair p into dst[0..31] (contiguous floats).
__device__ __forceinline__ void angular_compute(
    const float* __restrict__ vec, long P, long p,
    const float* __restrict__ etaA, const float* __restrict__ zeta,
    const float* __restrict__ shfA, const float* __restrict__ shfZ,
    float* dst)
{
    const float* a = vec + 3l * p;
    const float* b = vec + 3l * (P + p);
    float ax = a[0], ay = a[1], az = a[2];
    float bx = b[0], by = b[1], bz = b[2];

    float d0sq  = fmaf(ax, ax, fmaf(ay, ay, az * az));
    float d1sq  = fmaf(bx, bx, fmaf(by, by, bz * bz));
    float dot01 = fmaf(ax, bx, fmaf(ay, by, az * bz));
    float d0 = __builtin_amdgcn_sqrtf(d0sq);
    float d1 = __builtin_amdgcn_sqrtf(d1sq);

    float denom  = fmaxf(d0 * d1, 1e-10f);
    float cosang = __fdividef(dot01, denom);
    // angles = arccos(0.95*cosang); we only ever need cos/sin of it:
    float c = 0.95f * cosang;                                    // cos(angles)
    float s = __builtin_amdgcn_sqrtf(fmaxf(1.0f - c * c, 0.0f)); // sin(angles) >= 0 on [0,pi]

    const float kF = 3.14159265358979323846f / 3.5f;  // pi / CUTOFF
    float fc0 = fmaf(0.5f, __cosf(d0 * kF), 0.5f);
    float fc1 = fmaf(0.5f, __cosf(d1 * kF), 0.5f);
    float pref = 2.0f * fc0 * fc1;
    float dh   = 0.5f * (d0 + d1);

    float Zt = zeta[0];
    float negEtaLog2e = -etaA[0] * 1.4426950408889634f;  // -EtaA * log2(e)

    // f1_k scaled by pref:  ((1 + cos(a - ShfZ_k)) / 2)^Zeta * 2*fc0*fc1
    float g[4];
#pragma unroll
    for (int k = 0; k < 4; ++k) {
        float sz, cz;
        __sincosf(shfZ[k], &sz, &cz);
        float t  = 0.5f * (1.0f + c * cz + s * sz);   // in [0,1]
        float f1 = __builtin_exp2f(Zt * __log2f(t));  // t^Zeta via native v_log/v_exp
        g[k] = pref * f1;
    }

    // f2_j = exp(-EtaA * (dh - ShfA_j)^2), outer product into dst
#pragma unroll
    for (int j = 0; j < 8; ++j) {
        float u  = dh - shfA[j];
        float f2 = __builtin_exp2f(negEtaLog2e * u * u);
        v4f v;
        v.x = f2 * g[0];
        v.y = f2 * g[1];
        v.z = f2 * g[2];
        v.w = f2 * g[3];
        *(v4f*)(dst + 4 * j) = v;  // 16B-aligned (dst rows are 16B aligned)
    }
}

__global__ __launch_bounds__(256) void angular_main(
    const float* __restrict__ vec,
    const float* __restrict__ etaA, const float* __restrict__ zeta,
    const float* __restrict__ shfA, const float* __restrict__ shfZ,
    float* __restrict__ out, long P)
{
    __shared__ float lds[256 * LDS_ROW];  // 36 KB
    const int tid = threadIdx.x;
    const long p  = (long)blockIdx.x * 256 + tid;

    angular_compute(vec, P, p, etaA, zeta, shfA, shfZ, &lds[tid * LDS_ROW]);

    __syncthreads();

    // Coalesced tile writeout: lane-consecutive float4 chunks -> each wave store
    // instruction covers 512B of contiguous output.
    float* outBase = out + (long)blockIdx.x * (256l * 32l);
#pragma unroll
    for (int i = 0; i < 8; ++i) {
        int idx = i * 256 + tid;       // float4 index inside the block tile
        int lp  = idx >> 3;            // source pair row in LDS
        int kk  = (idx & 7) * 4;       // float offset within row
        float*       dst = outBase + 4l * idx;
        const float* src = &lds[lp * LDS_ROW + kk];
#if USE_ASYNC_STORE
        __builtin_amdgcn_global_store_async_from_lds_b128(
            (__attribute__((address_space(1))) v4i*)dst,
            (__attribute__((address_space(3))) v4i*)src,
            0, 0);
#else
        *(v4f*)dst = *(const v4f*)src;
#endif
    }
#if USE_ASYNC_STORE
#if __has_builtin(__builtin_amdgcn_s_wait_asynccnt)
    __builtin_amdgcn_s_wait_asynccnt(0);
#else
    asm volatile("s_wait_asynccnt 0" ::: "memory");
#endif
#endif
}

// Tail kernel for P not divisible by 256 (not hit for P = 4194304, kept for safety).
__global__ void angular_tail(
    const float* __restrict__ vec,
    const float* __restrict__ etaA, const float* __restrict__ zeta,
    const float* __restrict__ shfA, const float* __restrict__ shfZ,
    float* __restrict__ out, long P, long start)
{
    long p = start + (long)blockIdx.x * blockDim.x + threadIdx.x;
    if (p >= P) return;
    float buf[32] __attribute__((aligned(16)));
    angular_compute(vec, P, p, etaA, zeta, shfA, shfZ, buf);
#pragma unroll
    for (int i = 0; i < 32; ++i) out[p * 32 + i] = buf[i];
}

extern "C" void kernel_launch(void* const* d_in, const int* in_sizes, int n_in,
                              void* d_out, int out_size, void* d_ws, size_t ws_size,
                              hipStream_t stream)
{
    const float* vec  = (const float*)d_in[0];  // (2, P, 3)
    const float* etaA = (const float*)d_in[1];  // (1,)
    const float* zeta = (const float*)d_in[2];  // (1,)
    const float* shfA = (const float*)d_in[3];  // (8,)
    const float* shfZ = (const float*)d_in[4];  // (4,)
    float* out = (float*)d_out;                 // (P, 32)

    long P  = (long)in_sizes[0] / 6;
    long nb = P / 256;
    if (nb > 0) {
        angular_main<<<dim3((unsigned)nb), dim3(256), 0, stream>>>(
            vec, etaA, zeta, shfA, shfZ, out, P);
    }
    long done = nb * 256;
    long tail = P - done;
    if (tail > 0) {
        unsigned tb = (unsigned)((tail + 255) / 256);
        angular_tail<<<dim3(tb), dim3(256), 0, stream>>>(
            vec, etaA, zeta, shfA, shfZ, out, P, done);
    }
}